// OCGatherEnergyLoop_61237643706560
// MI455X (gfx1250) — compile-verified
//
#include <hip/hip_runtime.h>
#include <stdint.h>

// Segmented reduce + broadcast for OCGatherEnergyLoop.
// G (num_showers) is a compile-time constant in the reference (8192); at
// runtime it only exists as a device-side scalar, unreadable under graph
// capture, so we bake it in.
static constexpr int G = 8192;

// Native clang vector types: required by __builtin_nontemporal_load/store
// (HIP_vector_type structs are rejected), same 16B alignment -> b128 ops.
typedef __attribute__((ext_vector_type(4))) int vint4;
typedef __attribute__((ext_vector_type(4))) float vfloat4;
typedef __attribute__((ext_vector_type(2))) unsigned long long vull2;

// Explicit global address space so the atomics select the global_* forms.
#define GLOBAL_AS __attribute__((address_space(1)))
typedef GLOBAL_AS float gfloat;
typedef GLOBAL_AS unsigned long long gull;

__device__ __forceinline__ void accum_one(int s, int h, float e, float b, float c,
                                          gfloat* e_sum, gull* packed) {
  int idx = s + (h != 0 ? G : 0);
  // -> global_atomic_add_f32 (relaxed, agent scope)
  __hip_atomic_fetch_add(e_sum + idx, e, __ATOMIC_RELAXED,
                         __HIP_MEMORY_SCOPE_AGENT);
  // beta/corr are nonnegative finite floats: bit pattern orders like value.
  // Lexicographic (beta, corr) max == beta_max + max-corr-at-argmax.
  unsigned long long pk =
      ((unsigned long long)__float_as_uint(b) << 32) |
      (unsigned long long)__float_as_uint(c);
  // -> global_atomic_max_u64
  __hip_atomic_fetch_max(packed + idx, pk, __ATOMIC_RELAXED,
                         __HIP_MEMORY_SCOPE_AGENT);
}

// Hot kernel first in the file so the disasm snippet shows it.
__global__ void __launch_bounds__(256)
k_pass1_vec(const vint4* __restrict__ sid,
            const vint4* __restrict__ hit,
            const vfloat4* __restrict__ en,
            const vfloat4* __restrict__ be,
            const vfloat4* __restrict__ co,
            float* e_sum_p, unsigned long long* packed_p, int nvec) {
  gfloat* e_sum = (gfloat*)e_sum_p;
  gull* packed = (gull*)packed_p;
  int i = blockIdx.x * blockDim.x + threadIdx.x;
  if (i >= nvec) return;
  // Read-once streams: non-temporal b128 loads.
  vint4 s = __builtin_nontemporal_load(&sid[i]);
  vint4 h = __builtin_nontemporal_load(&hit[i]);
  vfloat4 e = __builtin_nontemporal_load(&en[i]);
  vfloat4 b = __builtin_nontemporal_load(&be[i]);
  vfloat4 c = __builtin_nontemporal_load(&co[i]);
  accum_one(s.x, h.x, e.x, b.x, c.x, e_sum, packed);
  accum_one(s.y, h.y, e.y, b.y, c.y, e_sum, packed);
  accum_one(s.z, h.z, e.z, b.z, c.z, e_sum, packed);
  accum_one(s.w, h.w, e.w, b.w, c.w, e_sum, packed);
}

__global__ void __launch_bounds__(256)
k_pass3_vec(const vint4* __restrict__ sid,
            const vfloat4* __restrict__ finals4,
            float* __restrict__ out, int nvec, int N) {
  int i = blockIdx.x * blockDim.x + threadIdx.x;
  if (i >= nvec) return;
  vint4 s = __builtin_nontemporal_load(&sid[i]);
  // Gather from the L2-resident per-shower table (keep temporal).
  vfloat4 a0 = finals4[2 * s.x], a1 = finals4[2 * s.x + 1];
  vfloat4 b0 = finals4[2 * s.y], b1 = finals4[2 * s.y + 1];
  vfloat4 c0 = finals4[2 * s.z], c1 = finals4[2 * s.z + 1];
  vfloat4 d0 = finals4[2 * s.w], d1 = finals4[2 * s.w + 1];
  const size_t nn = (size_t)N;
  // Explicit register transpose; channel-major, coalesced NT b128 stores.
  vfloat4 v;
  v = (vfloat4){a0.x, b0.x, c0.x, d0.x};
  __builtin_nontemporal_store(v, (vfloat4*)(out + 0 * nn) + i);
  v = (vfloat4){a0.y, b0.y, c0.y, d0.y};
  __builtin_nontemporal_store(v, (vfloat4*)(out + 1 * nn) + i);
  v = (vfloat4){a0.z, b0.z, c0.z, d0.z};
  __builtin_nontemporal_store(v, (vfloat4*)(out + 2 * nn) + i);
  v = (vfloat4){a0.w, b0.w, c0.w, d0.w};
  __builtin_nontemporal_store(v, (vfloat4*)(out + 3 * nn) + i);
  v = (vfloat4){a1.x, b1.x, c1.x, d1.x};
  __builtin_nontemporal_store(v, (vfloat4*)(out + 4 * nn) + i);
  v = (vfloat4){a1.y, b1.y, c1.y, d1.y};
  __builtin_nontemporal_store(v, (vfloat4*)(out + 5 * nn) + i);
  v = (vfloat4){a1.z, b1.z, c1.z, d1.z};
  __builtin_nontemporal_store(v, (vfloat4*)(out + 6 * nn) + i);
  v = (vfloat4){a1.w, b1.w, c1.w, d1.w};
  __builtin_nontemporal_store(v, (vfloat4*)(out + 7 * nn) + i);
}

__global__ void __launch_bounds__(256)
k_pass2(const float* __restrict__ e_sum,
        const unsigned long long* __restrict__ packed,
        vfloat4* __restrict__ finals4) {
  int g = blockIdx.x * blockDim.x + threadIdx.x;
  if (g >= G) return;
  float es_h = e_sum[g];
  float es_t = e_sum[G + g];
  float ch = __uint_as_float((unsigned)(packed[g] & 0xffffffffull));
  float ct = __uint_as_float((unsigned)(packed[G + g] & 0xffffffffull));
  // Empty segment: es==0 and packed==0 -> ec==0, matching the reference
  // (e_corr = corr_g * e_sum with e_sum == 0).
  float ec_h = ch * es_h;
  float ec_t = ct * es_t;
  vfloat4 lo, hi;
  lo.x = es_t;                            // tracks_raw
  lo.y = ec_t;                            // tracks_corrected
  lo.z = es_h;                            // hits_raw
  lo.w = ec_h;                            // hits_corrected
  hi.x = (es_t != 0.0f) ? es_t : es_h;    // prefer_tracks_raw
  hi.y = (ec_t != 0.0f) ? ec_t : ec_h;    // prefer_tracks_corrected
  hi.z = (es_h != 0.0f) ? es_h : es_t;    // prefer_hits_raw
  hi.w = (ec_h != 0.0f) ? ec_h : ec_t;    // prefer_hits_corrected
  finals4[2 * g + 0] = lo;
  finals4[2 * g + 1] = hi;
}

__global__ void __launch_bounds__(256)
k_init(vull2* __restrict__ p, int n) {
  int i = blockIdx.x * blockDim.x + threadIdx.x;
  if (i < n) {
    vull2 z = (vull2){0ull, 0ull};
    p[i] = z;
  }
}

__global__ void __launch_bounds__(256)
k_pass1_scalar(const int* __restrict__ sid,
               const int* __restrict__ hit,
               const float* __restrict__ en,
               const float* __restrict__ be,
               const float* __restrict__ co,
               float* e_sum_p, unsigned long long* packed_p,
               int start, int n) {
  gfloat* e_sum = (gfloat*)e_sum_p;
  gull* packed = (gull*)packed_p;
  int i = start + blockIdx.x * blockDim.x + threadIdx.x;
  if (i >= n) return;
  accum_one(sid[i], hit[i], en[i], be[i], co[i], e_sum, packed);
}

__global__ void __launch_bounds__(256)
k_pass3_scalar(const int* __restrict__ sid,
               const float* __restrict__ finals,
               float* __restrict__ out, int start, int N) {
  int i = start + blockIdx.x * blockDim.x + threadIdx.x;
  if (i >= N) return;
  int g = sid[i];
#pragma unroll
  for (int c = 0; c < 8; ++c)
    out[(size_t)c * (size_t)N + i] = finals[g * 8 + c];
}

extern "C" void kernel_launch(void* const* d_in, const int* in_sizes, int n_in,
                              void* d_out, int out_size, void* d_ws,
                              size_t ws_size, hipStream_t stream) {
  const int N = in_sizes[0];
  const int* sid = (const int*)d_in[0];
  const int* hit = (const int*)d_in[1];
  const float* en = (const float*)d_in[2];
  const float* be = (const float*)d_in[3];
  const float* co = (const float*)d_in[4];
  float* out = (float*)d_out;

  char* ws = (char*)d_ws;
  float* e_sum = (float*)ws;                                        // 2G floats
  unsigned long long* packed =
      (unsigned long long*)(ws + (size_t)2 * G * sizeof(float));    // 2G u64
  float* finals =
      (float*)(ws + (size_t)2 * G * sizeof(float) +
               (size_t)2 * G * sizeof(unsigned long long));          // 8G floats

  const int B = 256;

  // Zero accumulator region (e_sum + packed) every call.
  const int init_bytes = 2 * G * (int)sizeof(float) +
                         2 * G * (int)sizeof(unsigned long long);
  const int init_n = init_bytes / (int)sizeof(vull2);
  k_init<<<(init_n + B - 1) / B, B, 0, stream>>>((vull2*)ws, init_n);

  // Pass 1: segmented atomics (f32 add + u64 lexicographic max).
  const int nvec = N / 4;
  if (nvec > 0)
    k_pass1_vec<<<(nvec + B - 1) / B, B, 0, stream>>>(
        (const vint4*)sid, (const vint4*)hit, (const vfloat4*)en,
        (const vfloat4*)be, (const vfloat4*)co, e_sum, packed, nvec);
  if (N - nvec * 4 > 0)
    k_pass1_scalar<<<1, B, 0, stream>>>(sid, hit, en, be, co, e_sum, packed,
                                        nvec * 4, N);

  // Pass 2: finalize all 8 per-shower channels.
  k_pass2<<<(G + B - 1) / B, B, 0, stream>>>(e_sum, packed, (vfloat4*)finals);

  // Pass 3: broadcast per-shower channels to every element.
  if (nvec > 0 && (N % 4) == 0)
    k_pass3_vec<<<(nvec + B - 1) / B, B, 0, stream>>>(
        (const vint4*)sid, (const vfloat4*)finals, out, nvec, N);
  else
    k_pass3_scalar<<<(N + B - 1) / B, B, 0, stream>>>(sid, finals, out, 0, N);
}